// GCNWithJK_4320737100494
// MI455X (gfx1250) — compile-verified
//
#include <hip/hip_runtime.h>
#include <hip/hip_bf16.h>

// Problem constants (match reference)
#define GN 100000   // nodes  (= 3125 * 32, exact GEMM tiling)
#define GE 1600000  // edges
#define GH 64       // hidden / F_IN
#define GL 4        // layers
#define GG 512      // graphs (= 16 * 32, exact GEMM tiling)
#define GC 10       // classes
#define HCAT 256    // GL*GH

typedef float v2f __attribute__((ext_vector_type(2)));
typedef float v8f __attribute__((ext_vector_type(8)));

// ---------------- degree / norm precompute ----------------

__global__ void k_init_deg(float* __restrict__ deg, int n) {
  int i = blockIdx.x * blockDim.x + threadIdx.x;
  if (i < n) deg[i] = 1.0f;  // self-loop contribution
}

__global__ void k_count_deg(const int* __restrict__ dst, float* __restrict__ deg, int e) {
  int i = blockIdx.x * blockDim.x + threadIdx.x;
  if (i < e) atomicAdd(&deg[dst[i]], 1.0f);
}

__global__ void k_dinv(float* __restrict__ deg, int n) {
  int i = blockIdx.x * blockDim.x + threadIdx.x;
  if (i < n) deg[i] = rsqrtf(deg[i]);  // deg >= 1 always (self-loops)
}

__global__ void k_norm(const int* __restrict__ src, const int* __restrict__ dst,
                       const float* __restrict__ dinv, float* __restrict__ norm, int e) {
  int i = blockIdx.x * blockDim.x + threadIdx.x;
  if (i < e) norm[i] = dinv[src[i]] * dinv[dst[i]];
}

// ---------------- WMMA f32 GEMM:  Y[rows x 64] = X[rows x K] @ W[K x 64] ----------------
// Block = 256 threads = 8 waves; each wave computes one 16x16 tile.
// Block covers 32 rows x 64 cols. W is row-major [K][64].
// REQUIREMENT: rows % 32 == 0 (true for N=100000 and G=512) -> no predication,
// EXEC stays all-ones through every WMMA, no per-iteration cndmask.
template <int K, int LDA>
__global__ __launch_bounds__(256)
void k_gemm_wmma(const float* __restrict__ X,
                 const float* __restrict__ W,
                 float* __restrict__ Y) {
  const int lane  = threadIdx.x & 31;
  const int wave  = threadIdx.x >> 5;
  const int row0  = blockIdx.x * 32 + (wave >> 2) * 16;  // 2 row tiles
  const int col0  = (wave & 3) * 16;                     // 4 col tiles
  const int m     = row0 + (lane & 15);
  const int n     = col0 + (lane & 15);
  const int khalf = (lane >> 4) << 1;  // 0 (lanes 0-15) or 2 (lanes 16-31)

  const float* xrow = X + (size_t)m * LDA + khalf;
  const float* wcol = W + (size_t)khalf * GH + n;

  v8f acc = {};
#pragma unroll
  for (int kk = 0; kk < K; kk += 4) {
    v2f a, b;
    // A (16x4 f32): lane l: VGPR j holds X[m][kk + 2*(l/16) + j]  (contiguous pair -> b64 load)
    a.x = xrow[kk];
    a.y = xrow[kk + 1];
    // B (4x16 f32): lane l: VGPR j holds W[kk + 2*(l/16) + j][n]
    b.x = wcol[(size_t)kk * GH];
    b.y = wcol[(size_t)(kk + 1) * GH];
    acc = __builtin_amdgcn_wmma_f32_16x16x4_f32(
        /*neg_a=*/false, a, /*neg_b=*/false, b,
        /*c_mod=*/(short)0, acc, /*reuse_a=*/false, /*reuse_b=*/false);
  }

  // C/D (16x16 f32): VGPR r: M = r + 8*(lane/16), N = lane%16
  float* yout = Y + (size_t)(row0 + ((lane >> 4) << 3)) * GH + n;
#pragma unroll
  for (int r = 0; r < 8; ++r) {
    yout[(size_t)r * GH] = acc[r];
  }
}

// ---------------- message passing ----------------

// Initialize hcat layer slice with the self-loop message: dinv[n]^2 * t[n][f]
__global__ void k_selfloop(const float* __restrict__ t, const float* __restrict__ dinv,
                           float* __restrict__ hcat, int layer, int n) {
  int i = blockIdx.x * blockDim.x + threadIdx.x;
  if (i >= n * GH) return;
  int node = i >> 6, f = i & 63;
  float d = dinv[node];
  hcat[(size_t)node * HCAT + layer * GH + f] = t[i] * d * d;
}

// 16 consecutive threads handle one edge (4 floats each): contiguous gather + atomics
__global__ void k_scatter(const int* __restrict__ src, const int* __restrict__ dst,
                          const float* __restrict__ norm, const float* __restrict__ t,
                          float* __restrict__ hcat, int layer, int e) {
  long long i = (long long)blockIdx.x * blockDim.x + threadIdx.x;
  if (i >= (long long)e * 16) return;
  int eid   = (int)(i >> 4);
  int chunk = ((int)i & 15) << 2;
  int s = src[eid], d = dst[eid];
  float w = norm[eid];
  const float4 v = *(const float4*)(t + (size_t)s * GH + chunk);
  float* out = hcat + (size_t)d * HCAT + layer * GH + chunk;
  atomicAdd(out + 0, v.x * w);
  atomicAdd(out + 1, v.y * w);
  atomicAdd(out + 2, v.z * w);
  atomicAdd(out + 3, v.w * w);
}

__global__ void k_bias_relu(float* __restrict__ hcat, const float* __restrict__ bias,
                            int layer, int n) {
  int i = blockIdx.x * blockDim.x + threadIdx.x;
  if (i >= n * GH) return;
  int node = i >> 6, f = i & 63;
  float* p = hcat + (size_t)node * HCAT + layer * GH + f;
  float v = *p + bias[f];
  *p = v > 0.0f ? v : 0.0f;
}

// ---------------- pooling (post-ReLU values >= 0: int-bit max == float max, init 0
//                  also reproduces the reference's empty-graph -> 0 guard) ----------------

__global__ void k_pool_init(float* __restrict__ pooled, int total) {
  int i = blockIdx.x * blockDim.x + threadIdx.x;
  if (i < total) pooled[i] = 0.0f;
}

__global__ void k_pool_max(const float* __restrict__ hcat, const int* __restrict__ batch,
                           float* __restrict__ pooled, int n) {
  int i = blockIdx.x * blockDim.x + threadIdx.x;
  if (i >= n * HCAT) return;
  int node = i >> 8, f = i & 255;
  float v = hcat[i];
  atomicMax((int*)pooled + (size_t)batch[node] * HCAT + f, __float_as_int(v));
}

// ---------------- head ----------------

__global__ void k_head1(float* __restrict__ y, const float* __restrict__ b, int total) {
  int i = blockIdx.x * blockDim.x + threadIdx.x;
  if (i >= total) return;
  float v = y[i] + b[i & 63];
  y[i] = v > 0.0f ? v : 0.0f;
}

__global__ void k_head2(const float* __restrict__ h1, const float* __restrict__ W2,
                        const float* __restrict__ b2, float* __restrict__ out, int g) {
  int gi = blockIdx.x * blockDim.x + threadIdx.x;
  if (gi >= g) return;
  float logits[GC];
#pragma unroll
  for (int c = 0; c < GC; ++c) logits[c] = b2[c];
  const float* hr = h1 + (size_t)gi * GH;
  for (int k = 0; k < GH; ++k) {
    float hv = hr[k];
#pragma unroll
    for (int c = 0; c < GC; ++c) logits[c] = fmaf(hv, W2[k * GC + c], logits[c]);
  }
  float mx = logits[0];
#pragma unroll
  for (int c = 1; c < GC; ++c) mx = fmaxf(mx, logits[c]);
  float lse = 0.0f;
#pragma unroll
  for (int c = 0; c < GC; ++c) lse += expf(logits[c] - mx);
  lse = logf(lse);
#pragma unroll
  for (int c = 0; c < GC; ++c) out[(size_t)gi * GC + c] = logits[c] - mx - lse;
}

// ---------------- driver ----------------

extern "C" void kernel_launch(void* const* d_in, const int* in_sizes, int n_in,
                              void* d_out, int out_size, void* d_ws, size_t ws_size,
                              hipStream_t stream) {
  const float* x     = (const float*)d_in[0];
  const int*   ei    = (const int*)d_in[1];   // [2, E]
  const int*   batch = (const int*)d_in[2];   // [N]
  const float* Wl[GL] = {(const float*)d_in[3], (const float*)d_in[5],
                         (const float*)d_in[7], (const float*)d_in[9]};
  const float* bl[GL] = {(const float*)d_in[4], (const float*)d_in[6],
                         (const float*)d_in[8], (const float*)d_in[10]};
  const float* lin1W = (const float*)d_in[11];  // [256,64]
  const float* lin1b = (const float*)d_in[12];
  const float* lin2W = (const float*)d_in[13];  // [64,10]
  const float* lin2b = (const float*)d_in[14];
  float* out = (float*)d_out;                   // [G, C]

  const int* src = ei;       // edge_index[0]
  const int* dst = ei + GE;  // edge_index[1]

  // Workspace carve (~135 MB, fits 192 MB L2)
  float* ws     = (float*)d_ws;
  float* dinv   = ws;                          // N   (used as deg, then rsqrt in place)
  float* norm   = dinv + GN;                   // E
  float* hcat   = norm + GE;                   // N * 256
  float* t      = hcat + (size_t)GN * HCAT;    // N * 64 (per-layer GEMM output)
  float* pooled = t + (size_t)GN * GH;         // G * 256
  float* h1     = pooled + (size_t)GG * HCAT;  // G * 64

  dim3 B(256);

  // degree / norm
  k_init_deg <<<(GN + 255) / 256, B, 0, stream>>>(dinv, GN);
  k_count_deg<<<(GE + 255) / 256, B, 0, stream>>>(dst, dinv, GE);
  k_dinv     <<<(GN + 255) / 256, B, 0, stream>>>(dinv, GN);
  k_norm     <<<(GE + 255) / 256, B, 0, stream>>>(src, dst, dinv, norm, GE);

  // 4 GCN layers
  for (int l = 0; l < GL; ++l) {
    if (l == 0) {
      k_gemm_wmma<GH, GH><<<GN / 32, B, 0, stream>>>(x, Wl[0], t);
    } else {
      k_gemm_wmma<GH, HCAT><<<GN / 32, B, 0, stream>>>(hcat + (size_t)(l - 1) * GH, Wl[l], t);
    }
    k_selfloop <<<(GN * GH + 255) / 256, B, 0, stream>>>(t, dinv, hcat, l, GN);
    k_scatter  <<<(unsigned)(((long long)GE * 16 + 255) / 256), B, 0, stream>>>(
        src, dst, norm, t, hcat, l, GE);
    k_bias_relu<<<(GN * GH + 255) / 256, B, 0, stream>>>(hcat, bl[l], l, GN);
  }

  // global max pool over graphs
  k_pool_init<<<(GG * HCAT + 255) / 256, B, 0, stream>>>(pooled, GG * HCAT);
  k_pool_max <<<(unsigned)(((size_t)GN * HCAT + 255) / 256), B, 0, stream>>>(
      hcat, batch, pooled, GN);

  // head: relu(pooled @ lin1_W + b), then logits + log_softmax
  k_gemm_wmma<HCAT, HCAT><<<GG / 32, B, 0, stream>>>(pooled, lin1W, h1);
  k_head1    <<<(GG * GH + 255) / 256, B, 0, stream>>>(h1, lin1b, GG * GH);
  k_head2    <<<(GG + 63) / 64, dim3(64), 0, stream>>>(h1, lin2W, lin2b, out, GG);
}